// GCN_31413390803462
// MI455X (gfx1250) — compile-verified
//
#include <hip/hip_runtime.h>
#include <hip/hip_bf16.h>

#define N_NODES 50000
#define FEAT    128
#define NCLASS  40

typedef float v2f __attribute__((ext_vector_type(2)));
typedef float v8f __attribute__((ext_vector_type(8)));

// ---------------------------------------------------------------------------
// GEMM: H[50000x128] = X[50000x128] @ W[128x128], fp32 WMMA 16x16x4.
// Block = 128 threads (4 waves). Block b owns rows [b*16, b*16+16).
// Wave w owns cols [w*32, w*32+32) as two 16-wide N tiles.
// A layout (16x4 f32): lane L -> m = L&15, k-pair base = (L>>4)*2, vgpr j -> k+j.
// B layout (4x16 f32): lane L -> n = L&15, k = (L>>4)*2 + j  (mirror of A).
// C/D layout (16x16 f32): vgpr j -> row = (L>>4)*8 + j, col = L&15.
// ---------------------------------------------------------------------------
__global__ __launch_bounds__(128) void gcn_gemm_xw(const float* __restrict__ X,
                                                   const float* __restrict__ W,
                                                   float* __restrict__ H) {
    const int wave = threadIdx.x >> 5;
    const int lane = threadIdx.x & 31;
    const int m    = lane & 15;
    const int kh   = (lane >> 4) * 2;           // 0 or 2
    const int half = lane >> 4;                 // 0 or 1
    const int row_base = blockIdx.x * 16;
    const int n0 = wave * 32 + (lane & 15);     // first N tile column for this lane

    const float* __restrict__ xrow = X + (row_base + m) * FEAT;

    v8f acc0 = {};
    v8f acc1 = {};

    #pragma unroll 4
    for (int k0 = 0; k0 < FEAT; k0 += 4) {
        v2f a = *(const v2f*)(xrow + k0 + kh);          // 8B aligned: kh even, k0 %4==0
        v2f b0, b1;
        const float* __restrict__ wr0 = W + (k0 + kh) * FEAT;
        b0.x = wr0[n0];
        b0.y = wr0[FEAT + n0];
        b1.x = wr0[n0 + 16];
        b1.y = wr0[FEAT + n0 + 16];
        acc0 = __builtin_amdgcn_wmma_f32_16x16x4_f32(false, a, false, b0,
                                                     (short)0, acc0, false, false);
        acc1 = __builtin_amdgcn_wmma_f32_16x16x4_f32(false, a, false, b1,
                                                     (short)0, acc1, false, false);
    }

    #pragma unroll
    for (int j = 0; j < 8; ++j) {
        const int r = row_base + half * 8 + j;
        H[r * FEAT + n0]      = acc0[j];
        H[r * FEAT + n0 + 16] = acc1[j];
    }
}

// ---------------------------------------------------------------------------
// Zero AGG (float4 granularity).
// ---------------------------------------------------------------------------
__global__ __launch_bounds__(256) void gcn_zero4(float4* __restrict__ p, int n4) {
    int i = blockIdx.x * blockDim.x + threadIdx.x;
    if (i < n4) p[i] = make_float4(0.f, 0.f, 0.f, 0.f);
}

// ---------------------------------------------------------------------------
// Scatter-add: AGG[dst[e]] += H[src[e]]. One thread per (edge, 4 feats):
// one b128 load + 4 f32 atomics that resolve in L2 (working set L2-resident).
// ---------------------------------------------------------------------------
__global__ __launch_bounds__(256) void gcn_scatter(const float* __restrict__ H,
                                                   const int* __restrict__ src,
                                                   const int* __restrict__ dst,
                                                   float* __restrict__ AGG,
                                                   int n_edges) {
    const int t    = blockIdx.x * blockDim.x + threadIdx.x;
    const int edge = t >> 5;
    const int f4   = (t & 31) * 4;
    if (edge >= n_edges) return;
    const int s = src[edge];
    const int d = dst[edge];
    const float4 v = *(const float4*)(H + (long long)s * FEAT + f4);
    float* p = AGG + (long long)d * FEAT + f4;
    atomicAdd(p + 0, v.x);
    atomicAdd(p + 1, v.y);
    atomicAdd(p + 2, v.z);
    atomicAdd(p + 3, v.w);
}

// ---------------------------------------------------------------------------
// out = relu( (agg + b) / max(||agg + b||_2, eps) ), one wave32 per node.
// ---------------------------------------------------------------------------
__global__ __launch_bounds__(256) void gcn_norm_relu(const float* __restrict__ AGG,
                                                     const float* __restrict__ bias,
                                                     float* __restrict__ OUT,
                                                     int n_nodes) {
    const int wave = threadIdx.x >> 5;
    const int lane = threadIdx.x & 31;
    const int node = blockIdx.x * 8 + wave;
    if (node >= n_nodes) return;

    const float* __restrict__ row = AGG + (long long)node * FEAT;
    float v[4];
    float ss = 0.f;
    #pragma unroll
    for (int j = 0; j < 4; ++j) {
        const float t = row[lane + j * 32] + bias[lane + j * 32];
        v[j] = t;
        ss += t * t;
    }
    #pragma unroll
    for (int off = 16; off; off >>= 1) ss += __shfl_xor(ss, off, 32);

    const float scale = 1.0f / fmaxf(sqrtf(ss), 1e-12f);
    float* __restrict__ orow = OUT + (long long)node * FEAT;
    #pragma unroll
    for (int j = 0; j < 4; ++j) {
        const float t = v[j] * scale;
        orow[lane + j * 32] = t > 0.f ? t : 0.f;
    }
}

// ---------------------------------------------------------------------------
// Classifier + softmax: logits = x4 @ Wl + bl; probs = softmax(logits).
// One 64-thread block per node; x row staged in LDS.
// ---------------------------------------------------------------------------
__global__ __launch_bounds__(64) void gcn_classify(const float* __restrict__ X4,
                                                   const float* __restrict__ Wl,
                                                   const float* __restrict__ bl,
                                                   float* __restrict__ logits,
                                                   float* __restrict__ probs) {
    __shared__ float xs[FEAT];
    __shared__ float ls[NCLASS];
    const int node = blockIdx.x;
    const int t = threadIdx.x;

    xs[t]      = X4[node * FEAT + t];
    xs[t + 64] = X4[node * FEAT + t + 64];
    __syncthreads();

    if (t < NCLASS) {
        float acc = bl[t];
        #pragma unroll 8
        for (int k = 0; k < FEAT; ++k) acc += xs[k] * Wl[k * NCLASS + t];
        ls[t] = acc;
        logits[node * NCLASS + t] = acc;
    }
    __syncthreads();

    if (t < NCLASS) {
        float mx = -3.402823466e38f;
        #pragma unroll 8
        for (int c = 0; c < NCLASS; ++c) mx = fmaxf(mx, ls[c]);
        float sum = 0.f;
        #pragma unroll 8
        for (int c = 0; c < NCLASS; ++c) sum += __expf(ls[c] - mx);
        probs[node * NCLASS + t] = __expf(ls[t] - mx) / sum;
    }
}

// ---------------------------------------------------------------------------
// Driver. Inputs: x, edge_index, W1,b1, W2,b2, W3,b3, W4,b4, Wl, bl.
// Output: [logits(50000x40), probs(50000x40), x4(50000x128)] flat.
// ---------------------------------------------------------------------------
extern "C" void kernel_launch(void* const* d_in, const int* in_sizes, int n_in,
                              void* d_out, int out_size, void* d_ws, size_t ws_size,
                              hipStream_t stream) {
    (void)n_in; (void)out_size; (void)ws_size;

    const float* x  = (const float*)d_in[0];
    const int*   ei = (const int*)d_in[1];
    const int n_edges = in_sizes[1] / 2;
    const int* src = ei;            // edge_index row 0
    const int* dst = ei + n_edges;  // edge_index row 1

    const float* Wm[4] = { (const float*)d_in[2], (const float*)d_in[4],
                           (const float*)d_in[6], (const float*)d_in[8] };
    const float* bm[4] = { (const float*)d_in[3], (const float*)d_in[5],
                           (const float*)d_in[7], (const float*)d_in[9] };
    const float* Wl = (const float*)d_in[10];
    const float* bl = (const float*)d_in[11];

    float* out    = (float*)d_out;
    float* logits = out;                                   // 50000*40
    float* probs  = out + (size_t)N_NODES * NCLASS;        // 50000*40
    float* x4     = out + (size_t)2 * N_NODES * NCLASS;    // 50000*128

    const size_t mat = (size_t)N_NODES * FEAT;             // 6.4M floats
    float* H   = (float*)d_ws;
    float* AGG = H + mat;
    float* XB  = AGG + mat;

    const int gemm_blocks    = N_NODES / 16;               // 3125, exact
    const int zero_blocks    = (int)((mat / 4 + 255) / 256);
    const int scatter_blocks = (n_edges * 32 + 255) / 256;
    const int norm_blocks    = (N_NODES + 7) / 8;

    const float* cur = x;
    for (int l = 0; l < 4; ++l) {
        float* nxt = (l == 3) ? x4 : XB;
        gcn_gemm_xw <<<gemm_blocks,    128, 0, stream>>>(cur, Wm[l], H);
        gcn_zero4   <<<zero_blocks,    256, 0, stream>>>((float4*)AGG, (int)(mat / 4));
        gcn_scatter <<<scatter_blocks, 256, 0, stream>>>(H, src, dst, AGG, n_edges);
        gcn_norm_relu<<<norm_blocks,   256, 0, stream>>>(AGG, bm[l], nxt, N_NODES);
        cur = nxt;
    }
    gcn_classify<<<N_NODES, 64, 0, stream>>>(x4, Wl, bl, logits, probs);
}